// GINEncoder_72284299592043
// MI455X (gfx1250) — compile-verified
//
#include <hip/hip_runtime.h>

#define N_NODES_C 100000
#define N_EDGES_C 1600000
#define EMB 64
#define NGRAPH 512
#define BN_EPS 1e-5f

typedef __attribute__((ext_vector_type(2))) float v2f;
typedef __attribute__((ext_vector_type(8))) float v8f;

__device__ __forceinline__ void atomAddF(float* p, float v) {
    __hip_atomic_fetch_add(p, v, __ATOMIC_RELAXED, __HIP_MEMORY_SCOPE_AGENT);
}

// -------- Edge scatter-add: agg[dst] += h[src], 64 dims, 16 threads/edge ----
__global__ __launch_bounds__(256) void scatter_add_kernel(
    const int* __restrict__ src, const int* __restrict__ dst,
    const float* __restrict__ h, float* __restrict__ agg, int nedges) {
    long long idx = (long long)blockIdx.x * blockDim.x + threadIdx.x;
    int e = (int)(idx >> 4);
    if (e >= nedges) return;
    int c = ((int)idx & 15) * 4;
    int s = src[e], d = dst[e];
    const float4 v = *(const float4*)(h + (size_t)s * EMB + c);
    float* o = agg + (size_t)d * EMB + c;
    atomAddF(o + 0, v.x);
    atomAddF(o + 1, v.y);
    atomAddF(o + 2, v.z);
    atomAddF(o + 3, v.w);
}

// -------- Fused (A [+A2]) @ W  -> affine(bias,BN) -> ReLU, via f32 WMMA -----
// Wave computes 16 rows x NC cols. Block = 8 waves = 128 rows.
// A fragment (16x4 f32): lane<16 holds K={k0,k0+1}, lane>=16 holds K={k0+2,k0+3}
// B fragment (4x16 f32): VGPR v: row k0+v (lanes 0-15), row k0+v+2 (lanes 16-31)
// C/D (16x16 f32): VGPR r: M=r (lanes 0-15), M=r+8 (lanes 16-31)
//
// W is staged in LDS K-pair-interleaved: sW2[(k/2)*NC + col] = {W[k][col],
// W[k+1][col]} so each lane's B fragment is one aligned ds_load_b64.
template <int KD, int NC, bool ADD2>
__global__ __launch_bounds__(256) void gin_mlp_gemm(
    const float* __restrict__ A, const float* __restrict__ A2,
    const float* __restrict__ W,
    const float* __restrict__ bias, const float* __restrict__ bn_beta,
    const float* __restrict__ bn_gamma, const float* __restrict__ bn_mean,
    const float* __restrict__ bn_var,
    float* __restrict__ out, int nrows) {
    __shared__ float sW[KD * NC];
    {
        const float4* Wv = (const float4*)W;
        for (int i4 = threadIdx.x; i4 < KD * NC / 4; i4 += 256) {
            int i = i4 * 4;
            int k = i / NC;
            int col = i - k * NC;
            float4 w = Wv[i4];
            // pair-interleaved destination: ((k/2)*NC + col)*2 + (k&1)
            int base = ((k >> 1) * NC + col) * 2 + (k & 1);
            sW[base + 0] = w.x;
            sW[base + 2] = w.y;
            sW[base + 4] = w.z;
            sW[base + 6] = w.w;
        }
    }
    __syncthreads();

    const int lane = threadIdx.x & 31;
    const int wave = threadIdx.x >> 5;
    const int row0 = blockIdx.x * 128 + wave * 16;
    const int l16  = lane & 15;
    const int koff = (lane >> 4) * 2;
    const int phalf = lane >> 4;  // which K-pair within the 4-wide step

    int mrow = row0 + l16;
    if (mrow > nrows - 1) mrow = nrows - 1;  // clamp loads; stores are masked
    const float* arow  = A + (size_t)mrow * KD;
    const float* arow2 = ADD2 ? (A2 + (size_t)mrow * KD) : nullptr;

    const v2f* sW2 = (const v2f*)sW;  // (KD/2) x NC pairs

    constexpr int NT = NC / 16;
    v8f acc[NT] = {};

#pragma unroll
    for (int k0 = 0; k0 < KD; k0 += 4) {
        float2 a = *(const float2*)(arow + k0 + koff);
        if (ADD2) {
            float2 a2 = *(const float2*)(arow2 + k0 + koff);
            a.x += a2.x;
            a.y += a2.y;
        }
        v2f af;
        af.x = a.x;
        af.y = a.y;
        const int prow = ((k0 >> 1) + phalf) * NC + l16;
#pragma unroll
        for (int t = 0; t < NT; ++t) {
            v2f bf = sW2[prow + t * 16];
            acc[t] = __builtin_amdgcn_wmma_f32_16x16x4_f32(
                false, af, false, bf, (short)0, acc[t], false, false);
        }
    }

    const int rbase = row0 + (lane >> 4) * 8;
#pragma unroll
    for (int t = 0; t < NT; ++t) {
        int col = t * 16 + l16;
        float s  = bn_gamma[col] * rsqrtf(bn_var[col] + BN_EPS);
        float tt = (bias[col] - bn_mean[col]) * s + bn_beta[col];
#pragma unroll
        for (int r = 0; r < 8; ++r) {
            int row = rbase + r;
            if (row < nrows) {
                float v = acc[t][r] * s + tt;
                out[(size_t)row * NC + col] = v > 0.f ? v : 0.f;
            }
        }
    }
}

// -------- Graph mean-pool ---------------------------------------------------
__global__ __launch_bounds__(256) void pool_scatter(
    const float* __restrict__ h, const int* __restrict__ batch,
    float* __restrict__ psum, unsigned* __restrict__ pcnt) {
    long long idx = (long long)blockIdx.x * blockDim.x + threadIdx.x;
    int n = (int)(idx >> 4);
    if (n >= N_NODES_C) return;
    int c = ((int)idx & 15) * 4;
    int g = batch[n];
    const float4 v = *(const float4*)(h + (size_t)n * EMB + c);
    float* o = psum + (size_t)g * EMB + c;
    atomAddF(o + 0, v.x);
    atomAddF(o + 1, v.y);
    atomAddF(o + 2, v.z);
    atomAddF(o + 3, v.w);
    if (c == 0)
        __hip_atomic_fetch_add(&pcnt[g], 1u, __ATOMIC_RELAXED, __HIP_MEMORY_SCOPE_AGENT);
}

__global__ __launch_bounds__(256) void pool_final(
    const float* __restrict__ psum, const unsigned* __restrict__ pcnt,
    float* __restrict__ out) {
    int i = blockIdx.x * blockDim.x + threadIdx.x;
    if (i >= NGRAPH * EMB) return;
    unsigned c = pcnt[i >> 6];
    float denom = (c > 0u) ? (float)c : 1.0f;
    out[i] = psum[i] / denom;
}

// ---------------------------------------------------------------------------
extern "C" void kernel_launch(void* const* d_in, const int* in_sizes, int n_in,
                              void* d_out, int out_size, void* d_ws, size_t ws_size,
                              hipStream_t stream) {
    (void)in_sizes; (void)n_in; (void)out_size; (void)ws_size;

    // Inputs: x [N,64] f32; edge_index [2,E] i32; batch [N] i32; then params
    // flattened per layer in sorted-key order:
    // b1, b2, bn1{beta,gamma,mean,var}, bn_out{beta,gamma,mean,var}, w1, w2
    const float* x    = (const float*)d_in[0];
    const int*  eidx  = (const int*)d_in[1];
    const int*  batch = (const int*)d_in[2];
    const int*  src   = eidx;
    const int*  dst   = eidx + N_EDGES_C;

    struct Layer {
        const float *b1, *b2;
        const float *bn1b, *bn1g, *bn1m, *bn1v;
        const float *bnob, *bnog, *bnom, *bnov;
        const float *w1, *w2;
    } L[3];
    int p = 3;
    for (int l = 0; l < 3; ++l) {
        L[l].b1   = (const float*)d_in[p + 0];
        L[l].b2   = (const float*)d_in[p + 1];
        L[l].bn1b = (const float*)d_in[p + 2];
        L[l].bn1g = (const float*)d_in[p + 3];
        L[l].bn1m = (const float*)d_in[p + 4];
        L[l].bn1v = (const float*)d_in[p + 5];
        L[l].bnob = (const float*)d_in[p + 6];
        L[l].bnog = (const float*)d_in[p + 7];
        L[l].bnom = (const float*)d_in[p + 8];
        L[l].bnov = (const float*)d_in[p + 9];
        L[l].w1   = (const float*)d_in[p + 10];
        L[l].w2   = (const float*)d_in[p + 11];
        p += 12;
    }

    // Workspace layout
    float*    hbuf = (float*)d_ws;                        // N*64
    float*    agg  = hbuf + (size_t)N_NODES_C * EMB;      // N*64
    float*    z1   = agg + (size_t)N_NODES_C * EMB;       // N*128
    float*    psum = z1 + (size_t)N_NODES_C * 2 * EMB;    // 512*64
    unsigned* pcnt = (unsigned*)(psum + NGRAPH * EMB);    // 512

    const int scatterBlocks = (N_EDGES_C * 16 + 255) / 256;
    const int gemmBlocks    = (N_NODES_C + 127) / 128;

    const float* cur = x;
    for (int l = 0; l < 3; ++l) {
        hipMemsetAsync(agg, 0, (size_t)N_NODES_C * EMB * sizeof(float), stream);
        scatter_add_kernel<<<scatterBlocks, 256, 0, stream>>>(src, dst, cur, agg, N_EDGES_C);
        if (l == 0) {
            gin_mlp_gemm<64, 128, true><<<gemmBlocks, 256, 0, stream>>>(
                cur, agg, L[0].w1, L[0].b1, L[0].bn1b, L[0].bn1g, L[0].bn1m,
                L[0].bn1v, z1, N_NODES_C);
            gin_mlp_gemm<128, 64, false><<<gemmBlocks, 256, 0, stream>>>(
                z1, nullptr, L[0].w2, L[0].b2, L[0].bnob, L[0].bnog, L[0].bnom,
                L[0].bnov, hbuf, N_NODES_C);
        } else {
            gin_mlp_gemm<64, 64, true><<<gemmBlocks, 256, 0, stream>>>(
                cur, agg, L[l].w1, L[l].b1, L[l].bn1b, L[l].bn1g, L[l].bn1m,
                L[l].bn1v, z1, N_NODES_C);
            gin_mlp_gemm<64, 64, false><<<gemmBlocks, 256, 0, stream>>>(
                z1, nullptr, L[l].w2, L[l].b2, L[l].bnob, L[l].bnog, L[l].bnom,
                L[l].bnov, hbuf, N_NODES_C);
        }
        cur = hbuf;
    }

    hipMemsetAsync(psum, 0,
                   (size_t)NGRAPH * EMB * sizeof(float) + NGRAPH * sizeof(unsigned),
                   stream);
    pool_scatter<<<(N_NODES_C * 16 + 255) / 256, 256, 0, stream>>>(hbuf, batch, psum, pcnt);
    pool_final<<<(NGRAPH * EMB + 255) / 256, 256, 0, stream>>>(psum, pcnt, (float*)d_out);
}